// AttnGate_4879082848825
// MI455X (gfx1250) — compile-verified
//
#include <hip/hip_runtime.h>

typedef __attribute__((ext_vector_type(2))) float v2f;
typedef __attribute__((ext_vector_type(8))) float v8f;

namespace {
constexpr int B_  = 4;
constexpr int S_  = 65536;
constexpr int HK  = 8;
constexpr int D_  = 128;
constexpr int BLK = 64;
constexpr int NB  = S_ / BLK;   // 1024
constexpr int DG  = 128;
constexpr int G_  = 4;
constexpr float EPS   = 1e-6f;
constexpr float SCALE = 0.08838834764831845f;  // 1/sqrt(128)
constexpr int KP_STRIDE = 260;  // 256 + 4 pad: banks differ by 4 per row
constexpr int CT_STRIDE = 132;  // 128 + 4 pad
}

// ---------------------------------------------------------------------------
// Gate query: qp[b,h,o] = sum_{g,i} q[b,0,h*4+g,i] * Wq[h,g,i,o]; rmsnorm; rope
// ---------------------------------------------------------------------------
__global__ __launch_bounds__(128)
void qgate_kernel(const float* __restrict__ q, const float* __restrict__ Wq,
                  const float* __restrict__ qnw, const float* __restrict__ cq,
                  const float* __restrict__ sq, float* __restrict__ qv) {
  const int h = blockIdx.x, b = blockIdx.y;
  const int o = threadIdx.x;  // 0..127 (Dg)
  float acc = 0.f;
  for (int gg = 0; gg < G_; ++gg) {
    const float* qrow = q + ((size_t)b * (HK * G_) + h * G_ + gg) * D_;
    const float* wcol = Wq + ((size_t)(h * G_ + gg) * D_) * DG + o;
    for (int i = 0; i < D_; ++i) acc += qrow[i] * wcol[(size_t)i * DG];
  }
  __shared__ float red[128];
  __shared__ float ybuf[128];
  red[o] = acc * acc;
  __syncthreads();
  for (int st = 64; st > 0; st >>= 1) {
    if (o < st) red[o] += red[o + st];
    __syncthreads();
  }
  const float rn = rsqrtf(red[0] / (float)DG + EPS);
  const float y = acc * rn * qnw[o];
  ybuf[o] = y;
  __syncthreads();
  const float rot = (o < DG / 2) ? -ybuf[o + DG / 2] : ybuf[o - DG / 2];
  qv[((size_t)b * HK + h) * DG + o] =
      y * cq[(size_t)b * DG + o] + rot * sq[(size_t)b * DG + o];
}

// ---------------------------------------------------------------------------
// Fused: pool 16 key-blocks -> LDS, WMMA-project (K=256 -> Dg=128),
// rmsnorm + rope + dot(qv) -> masked logits.  One global pass over K.
// ---------------------------------------------------------------------------
__global__ __launch_bounds__(256)
void kgate_kernel(const float* __restrict__ k, const float* __restrict__ Wk,
                  const float* __restrict__ knw, const float* __restrict__ ck,
                  const float* __restrict__ sk, const float* __restrict__ qv,
                  const int* __restrict__ amask, float* __restrict__ logits) {
  const int tm = blockIdx.x;  // 16-row tile of key-blocks (0..63)
  const int h  = blockIdx.y;
  const int b  = blockIdx.z;
  const int t = threadIdx.x;
  const int w = t >> 5, lane = t & 31;
  const int half = lane >> 4, li = lane & 15;

  __shared__ float kpool[16 * KP_STRIDE];  // [row][mean(128) | max(128)]
  __shared__ float tileC[16 * CT_STRIDE];  // k_c tile
  __shared__ float ynrm[16 * CT_STRIDE];   // rmsnorm'd k_c

  // ---- Phase 1: pool (mean + max over 64 positions), coalesced float4 ----
  for (int rr = 0; rr < 2; ++rr) {
    const int row = w * 2 + rr;            // 0..15
    const int sblk = tm * 16 + row;        // global key-block index
    const float* base =
        k + (((size_t)b * S_ + (size_t)sblk * BLK) * HK + h) * D_ + lane * 4;
    float4 sum = make_float4(0.f, 0.f, 0.f, 0.f);
    float4 mx = make_float4(-__builtin_inff(), -__builtin_inff(),
                            -__builtin_inff(), -__builtin_inff());
    for (int r = 0; r < BLK; ++r) {
      const float4 v = *reinterpret_cast<const float4*>(base + (size_t)r * (HK * D_));
      sum.x += v.x; sum.y += v.y; sum.z += v.z; sum.w += v.w;
      mx.x = fmaxf(mx.x, v.x); mx.y = fmaxf(mx.y, v.y);
      mx.z = fmaxf(mx.z, v.z); mx.w = fmaxf(mx.w, v.w);
    }
    const float inv = 1.f / (float)BLK;
    *reinterpret_cast<float4*>(&kpool[row * KP_STRIDE + lane * 4]) =
        make_float4(sum.x * inv, sum.y * inv, sum.z * inv, sum.w * inv);
    *reinterpret_cast<float4*>(&kpool[row * KP_STRIDE + 128 + lane * 4]) = mx;
  }
  __syncthreads();

  // ---- Phase 2: C(16x16) = A(16xK from LDS) * Wk(Kx16), f32 WMMA ----
  // A layout (16x4 f32): lanes 0-15 m=lane {v0=K0,v1=K1}; lanes 16-31 {K2,K3}
  const int n = w * 16 + li;  // this wave's Dg column group
  const float* wkh = Wk + (size_t)h * (2 * D_) * DG;
  v8f c = {};
  for (int k0 = 0; k0 < 2 * D_; k0 += 4) {
    const int kk = k0 + half * 2;
    const v2f a = *reinterpret_cast<const v2f*>(&kpool[li * KP_STRIDE + kk]);
    v2f bb;
    bb[0] = wkh[(size_t)kk * DG + n];
    bb[1] = wkh[(size_t)(kk + 1) * DG + n];
    c = __builtin_amdgcn_wmma_f32_16x16x4_f32(false, a, false, bb, (short)0, c,
                                              false, false);
  }
#pragma unroll
  for (int r = 0; r < 8; ++r)  // C: vgpr r, lanes0-15 M=r; lanes16-31 M=r+8
    tileC[(r + half * 8) * CT_STRIDE + w * 16 + li] = c[r];
  __syncthreads();

  // ---- Phase 3: per-row rmsnorm (16 lanes per row, 8 cols each) ----
  const int row = t >> 4;  // 0..15
  const int sub = t & 15;
  const int s = tm * 16 + row;  // global key-block
  float ss = 0.f;
#pragma unroll
  for (int j = 0; j < 8; ++j) {
    const float x = tileC[row * CT_STRIDE + sub * 8 + j];
    ss += x * x;
  }
  ss += __shfl_xor(ss, 8);
  ss += __shfl_xor(ss, 4);
  ss += __shfl_xor(ss, 2);
  ss += __shfl_xor(ss, 1);
  const float rn = rsqrtf(ss / (float)DG + EPS);
#pragma unroll
  for (int j = 0; j < 8; ++j) {
    const int o = sub * 8 + j;
    ynrm[row * CT_STRIDE + o] = tileC[row * CT_STRIDE + o] * rn * knw[o];
  }
  __syncthreads();

  // ---- Phase 4: rope + dot with gate query -> masked logit ----
  const float* ckr = ck + ((size_t)b * NB + s) * DG;
  const float* skr = sk + ((size_t)b * NB + s) * DG;
  const float* qvr = qv + ((size_t)b * HK + h) * DG;
  float pd = 0.f;
#pragma unroll
  for (int j = 0; j < 8; ++j) {
    const int o = sub * 8 + j;
    const float y = ynrm[row * CT_STRIDE + o];
    const float rot = (o < DG / 2) ? -ynrm[row * CT_STRIDE + o + DG / 2]
                                   :  ynrm[row * CT_STRIDE + o - DG / 2];
    pd += (y * ckr[o] + rot * skr[o]) * qvr[o];
  }
  pd += __shfl_xor(pd, 8);
  pd += __shfl_xor(pd, 4);
  pd += __shfl_xor(pd, 2);
  pd += __shfl_xor(pd, 1);
  if (sub == 0) {
    const size_t mi = ((size_t)b * HK + h) * NB + s;
    logits[mi] = amask[mi] ? pd * SCALE : -1e20f;
  }
}

// ---------------------------------------------------------------------------
// Exact top-k via rank counting (softmax is monotonic -> operate on logits).
// Tie-break = lower index first, matching jax.lax.top_k.
// ---------------------------------------------------------------------------
__global__ __launch_bounds__(1024)
void topk_kernel(const float* __restrict__ logits, const int* __restrict__ amask,
                 const int* __restrict__ budget_p, float* __restrict__ out) {
  const int bh = blockIdx.x;   // b*HK + h
  const int s = threadIdx.x;   // key-block 0..1023
  __shared__ float vals[NB];
  const float v = logits[(size_t)bh * NB + s];
  vals[s] = v;
  __syncthreads();
  const int budget = *budget_p;
  int rank = 0;
  for (int j = 0; j < NB; ++j) {
    const float u = vals[j];  // same address across lanes -> LDS broadcast
    rank += (int)((u > v) || (u == v && j < s));
  }
  float o = (rank < budget && amask[(size_t)bh * NB + s] != 0) ? 1.0f : 0.0f;
  if (s == NB - 1) o = 1.0f;  // mask[:, :, -1] = True
  out[(size_t)bh * NB + s] = o;
}

extern "C" void kernel_launch(void* const* d_in, const int* in_sizes, int n_in,
                              void* d_out, int out_size, void* d_ws, size_t ws_size,
                              hipStream_t stream) {
  const float* k    = (const float*)d_in[0];
  const float* q    = (const float*)d_in[1];
  const float* Wq   = (const float*)d_in[2];
  const float* Wk   = (const float*)d_in[3];
  const float* qnw  = (const float*)d_in[4];
  const float* knw  = (const float*)d_in[5];
  const float* cq   = (const float*)d_in[6];
  const float* sq   = (const float*)d_in[7];
  const float* ck   = (const float*)d_in[8];
  const float* sk   = (const float*)d_in[9];
  const int* amask  = (const int*)d_in[10];
  const int* budget = (const int*)d_in[11];

  float* qv     = (float*)d_ws;                 // B*HK*DG   = 4096 floats
  float* logits = qv + (size_t)B_ * HK * DG;    // B*HK*NB   = 32768 floats
  float* out    = (float*)d_out;                // (B,HK,NB) booleans as float

  qgate_kernel<<<dim3(HK, B_), 128, 0, stream>>>(q, Wq, qnw, cq, sq, qv);
  kgate_kernel<<<dim3(NB / 16, HK, B_), 256, 0, stream>>>(k, Wk, knw, ck, sk,
                                                          qv, amask, logits);
  topk_kernel<<<dim3(B_ * HK), NB, 0, stream>>>(logits, amask, budget, out);
}